// ALIGNNLike_2465311228415
// MI455X (gfx1250) — compile-verified
//
#include <hip/hip_runtime.h>
#include <hip/hip_bf16.h>

#define H 128

typedef float v2f __attribute__((ext_vector_type(2)));
typedef float v8f __attribute__((ext_vector_type(8)));

static inline unsigned cdivu(long long a, int b) { return (unsigned)((a + b - 1) / b); }

// ---------------------------------------------------------------- degrees ---
__global__ void k_deg_accum(const int* __restrict__ src, const int* __restrict__ dst,
                            float* __restrict__ dout, float* __restrict__ din, int ne) {
  int i = blockIdx.x * blockDim.x + threadIdx.x;
  if (i < ne) {
    atomicAdd(&dout[src[i]], 1.0f);
    atomicAdd(&din[dst[i]], 1.0f);
  }
}

__global__ void k_inv_rsqrt(float* __restrict__ v, int n) {
  int i = blockIdx.x * blockDim.x + threadIdx.x;
  if (i < n) {
    float d = v[i];
    d = d < 1.0f ? 1.0f : d;
    v[i] = rsqrtf(d);
  }
}

// ------------------------------------------------------------ featurizers ---
__global__ void k_embed(const int* __restrict__ z, const float* __restrict__ emb,
                        float* __restrict__ h, int n) {
  long long t = (long long)blockIdx.x * blockDim.x + threadIdx.x;
  long long node = t >> 5;
  int c4 = (int)(t & 31);
  if (node < n) {
    const float4* s = (const float4*)(emb + (long long)z[node] * H);
    ((float4*)(h + node * H))[c4] = s[c4];
  }
}

__global__ void k_edge_init(const float* __restrict__ d, const float* __restrict__ w,
                            const float* __restrict__ b, float* __restrict__ e, int ne) {
  long long t = (long long)blockIdx.x * blockDim.x + threadIdx.x;
  long long ed = t >> 5;
  int c4 = (int)(t & 31);
  if (ed < ne) {
    float dv = d[ed];
    float4 wv = ((const float4*)w)[c4];
    float4 bv = ((const float4*)b)[c4];
    float4 r;
    r.x = dv * wv.x + bv.x;
    r.y = dv * wv.y + bv.y;
    r.z = dv * wv.z + bv.z;
    r.w = dv * wv.w + bv.w;
    ((float4*)(e + ed * H))[c4] = r;
  }
}

// --------------------------------------------- fused gather-scale-scatter ---
// agg[dst] += h[src] * deg_out^-1/2[src]   (one thread per edge x float4)
__global__ void k_scatter(const float* __restrict__ hsrc, const int* __restrict__ src,
                          const int* __restrict__ dst, const float* __restrict__ inv_out,
                          float* __restrict__ agg, int ne) {
  long long t = (long long)blockIdx.x * blockDim.x + threadIdx.x;
  long long ed = t >> 5;
  int c4 = (int)(t & 31);
  if (ed < ne) {
    int s = src[ed];
    int dd = dst[ed];
    float sc = inv_out[s];
    float4 v = ((const float4*)(hsrc + (long long)s * H))[c4];
    float* o = agg + (long long)dd * H + c4 * 4;
    atomicAdd(o + 0, v.x * sc);
    atomicAdd(o + 1, v.y * sc);
    atomicAdd(o + 2, v.z * sc);
    atomicAdd(o + 3, v.w * sc);
  }
}

// ----------------------------------------------------- WMMA fp32 GEMM -------
// out[r, :] = relu( (A[r, :] * inv_in[r]) @ W + bias ),  A:[nrows,128], W:[128,128]
// Block = 256 threads = 8 waves; block covers 16 rows, each wave one 16x16 tile.
// K loop: 32 x V_WMMA_F32_16X16X4_F32 with C-chaining.
#define AS_STRIDE 132
__global__ __launch_bounds__(256) void k_gemm_bias_relu(
    const float* __restrict__ A, const float* __restrict__ inv_in,
    const float* __restrict__ W, const float* __restrict__ bias,
    float* __restrict__ out, int nrows) {
  __shared__ __align__(16) float As[16 * AS_STRIDE];

  const int row0 = blockIdx.x * 16;
  const int tid = threadIdx.x;

  // Stage the 16x128 A tile (scaled by deg_in^-1/2) into LDS.
  for (int j = 0; j < 8; ++j) {
    int idx = tid + j * 256;      // 0..2047
    int r = idx >> 7;             // 0..15
    int c = idx & 127;            // 0..127
    int row = row0 + r;
    float v = 0.0f;
    if (row < nrows) v = A[(long long)row * H + c] * inv_in[row];
    As[r * AS_STRIDE + c] = v;
  }
  __syncthreads();

  const int wave = tid >> 5;
  const int lane = tid & 31;
  const int n0 = wave * 16;            // output-column tile
  const int m = lane & 15;             // A row within tile (both lane halves)
  const int khalf = (lane >> 4) << 1;  // K offset: 0 for lanes 0-15, 2 for 16-31
  const int nn = n0 + (lane & 15);     // B/C column

  v8f acc = {};
  for (int k = 0; k < H; k += 4) {
    // A 16x4 fragment: a.x = A[m][k+khalf], a.y = A[m][k+khalf+1]
    v2f a = *(const v2f*)&As[m * AS_STRIDE + k + khalf];
    // B 4x16 fragment: b.x = W[k+khalf][nn], b.y = W[k+khalf+1][nn]
    v2f b;
    b.x = W[(k + khalf) * H + nn];
    b.y = W[(k + khalf + 1) * H + nn];
    acc = __builtin_amdgcn_wmma_f32_16x16x4_f32(
        /*neg_a=*/false, a, /*neg_b=*/false, b,
        /*c_mod=*/(short)0, acc, /*reuse_a=*/false, /*reuse_b=*/false);
  }

  // C/D layout: VGPR r -> M=r (lanes 0-15) or M=r+8 (lanes 16-31), N=lane&15.
  const float bv = bias[nn];
  const int mhi = (lane >> 4) * 8;
  for (int r = 0; r < 8; ++r) {
    int row = row0 + r + mhi;
    if (row < nrows) {
      float v = acc[r] + bv;
      out[(long long)row * H + nn] = v > 0.0f ? v : 0.0f;
    }
  }
}

// -------------------------------------------------------------- pooling -----
__global__ void k_pool_accum(const float* __restrict__ h, const int* __restrict__ seg,
                             float* __restrict__ sums, float* __restrict__ cnt, int n) {
  long long t = (long long)blockIdx.x * blockDim.x + threadIdx.x;
  long long node = t >> 5;
  int c4 = (int)(t & 31);
  if (node < n) {
    int g = seg[node];
    float4 v = ((const float4*)(h + node * H))[c4];
    float* o = sums + (long long)g * H + c4 * 4;
    atomicAdd(o + 0, v.x);
    atomicAdd(o + 1, v.y);
    atomicAdd(o + 2, v.z);
    atomicAdd(o + 3, v.w);
    if (c4 == 0) atomicAdd(&cnt[g], 1.0f);
  }
}

// -------------------------------------------------------------- readout -----
// y[b] = silu(concat(hg,he) @ r1w + r1b) @ r2w + r2b   (one block per graph)
__global__ __launch_bounds__(128) void k_readout(
    const float* __restrict__ hg, const float* __restrict__ he,
    const float* __restrict__ cn, const float* __restrict__ ce,
    const float* __restrict__ r1w, const float* __restrict__ r1b,
    const float* __restrict__ r2w, const float* __restrict__ r2b,
    float* __restrict__ out) {
  __shared__ float x[2 * H];
  __shared__ float red[H];
  int b = blockIdx.x;
  int t = threadIdx.x;
  float c1 = cn[b]; c1 = c1 < 1.0f ? 1.0f : c1;
  float c2 = ce[b]; c2 = c2 < 1.0f ? 1.0f : c2;
  x[t] = hg[b * H + t] / c1;
  x[H + t] = he[b * H + t] / c2;
  __syncthreads();
  float acc = r1b[t];
  for (int k = 0; k < 2 * H; ++k) acc += x[k] * r1w[k * H + t];
  float s = acc / (1.0f + __expf(-acc));  // silu
  red[t] = s * r2w[t];
  __syncthreads();
  for (int off = H / 2; off > 0; off >>= 1) {
    if (t < off) red[t] += red[t + off];
    __syncthreads();
  }
  if (t == 0) out[b] = red[0] + r2b[0];
}

// ---------------------------------------------------------------------------
extern "C" void kernel_launch(void* const* d_in, const int* in_sizes, int n_in,
                              void* d_out, int out_size, void* d_ws, size_t ws_size,
                              hipStream_t stream) {
  const int*   z    = (const int*)  d_in[0];
  const float* d    = (const float*)d_in[1];
  const int*   src  = (const int*)  d_in[2];
  const int*   dst  = (const int*)  d_in[3];
  const int*   lsrc = (const int*)  d_in[4];
  const int*   ldst = (const int*)  d_in[5];
  const int*   ngr  = (const int*)  d_in[6];
  const int*   egr  = (const int*)  d_in[7];
  const float* emb  = (const float*)d_in[8];
  const float* ep_w = (const float*)d_in[9];
  const float* ep_b = (const float*)d_in[10];
  const float* gW   = (const float*)d_in[11];
  const float* gb   = (const float*)d_in[12];
  const float* lgW  = (const float*)d_in[13];
  const float* lgb  = (const float*)d_in[14];
  const float* r1w  = (const float*)d_in[15];
  const float* r1b  = (const float*)d_in[16];
  const float* r2w  = (const float*)d_in[17];
  const float* r2b  = (const float*)d_in[18];
  float* out = (float*)d_out;

  const int N  = in_sizes[0];
  const int E  = in_sizes[2];
  const int E2 = in_sizes[4];
  const int B  = out_size;
  const int L  = in_sizes[12] / H;  // gb is (L,H)

  // ------------------------------------------------ workspace layout (f32)
  float* ws = (float*)d_ws;
  size_t o = 0;
  float* h    = ws + o; o += (size_t)N * H;
  float* agg  = ws + o; o += (size_t)N * H;
  float* ef   = ws + o; o += (size_t)E * H;
  float* eagg = ws + o; o += (size_t)E * H;
  float* invs = ws + o; o += 2 * (size_t)N + 2 * (size_t)E;
  float* inv_out_n = invs;
  float* inv_in_n  = invs + N;
  float* inv_out_e = invs + 2 * (size_t)N;
  float* inv_in_e  = invs + 2 * (size_t)N + E;
  float* pool = ws + o; o += 2 * (size_t)B * H + 2 * (size_t)B;
  float* hg = pool;
  float* he = pool + (size_t)B * H;
  float* cn = pool + 2 * (size_t)B * H;
  float* ce = pool + 2 * (size_t)B * H + B;

  const int T = 256;

  // ------------------------------------------------ degrees -> rsqrt(max(.,1))
  hipMemsetAsync(invs, 0, sizeof(float) * (2 * (size_t)N + 2 * (size_t)E), stream);
  k_deg_accum<<<cdivu(E, T), T, 0, stream>>>(src, dst, inv_out_n, inv_in_n, E);
  k_deg_accum<<<cdivu(E2, T), T, 0, stream>>>(lsrc, ldst, inv_out_e, inv_in_e, E2);
  k_inv_rsqrt<<<cdivu(2LL * N + 2LL * E, T), T, 0, stream>>>(invs, 2 * N + 2 * E);

  // ------------------------------------------------ input features
  k_embed<<<cdivu((long long)N * 32, T), T, 0, stream>>>(z, emb, h, N);
  k_edge_init<<<cdivu((long long)E * 32, T), T, 0, stream>>>(d, ep_w, ep_b, ef, E);

  // ------------------------------------------------ node-graph conv layers
  for (int l = 0; l < L; ++l) {
    hipMemsetAsync(agg, 0, sizeof(float) * (size_t)N * H, stream);
    k_scatter<<<cdivu((long long)E * 32, T), T, 0, stream>>>(h, src, dst, inv_out_n, agg, E);
    k_gemm_bias_relu<<<cdivu(N, 16), T, 0, stream>>>(
        agg, inv_in_n, gW + (size_t)l * H * H, gb + (size_t)l * H, h, N);
  }

  // ------------------------------------------------ line-graph conv layers
  for (int l = 0; l < L; ++l) {
    hipMemsetAsync(eagg, 0, sizeof(float) * (size_t)E * H, stream);
    k_scatter<<<cdivu((long long)E2 * 32, T), T, 0, stream>>>(ef, lsrc, ldst, inv_out_e, eagg, E2);
    k_gemm_bias_relu<<<cdivu(E, 16), T, 0, stream>>>(
        eagg, inv_in_e, lgW + (size_t)l * H * H, lgb + (size_t)l * H, ef, E);
  }

  // ------------------------------------------------ pooling + readout
  hipMemsetAsync(pool, 0, sizeof(float) * (2 * (size_t)B * H + 2 * (size_t)B), stream);
  k_pool_accum<<<cdivu((long long)N * 32, T), T, 0, stream>>>(h, ngr, hg, cn, N);
  k_pool_accum<<<cdivu((long long)E * 32, T), T, 0, stream>>>(ef, egr, he, ce, E);
  k_readout<<<B, H, 0, stream>>>(hg, he, cn, ce, r1w, r1b, r2w, r2b, out);
}